// RNN_69501160784284
// MI455X (gfx1250) — compile-verified
//
#include <hip/hip_runtime.h>
#include <math.h>
#include <stdint.h>

// ---------------- CDNA5 fused RNN: x@W_ih^T + h@W_hh^T + tanh + FC head ----
// Grid: 128 blocks x 128 threads (4 wave32 per block). Each block owns 16
// batch rows for the whole 512-step recurrence; each wave owns a 16-column
// tile of the 64-wide hidden state. All matmuls use V_WMMA_F32_16X16X4_F32
// (fp32 in / fp32 acc -> exact-precision match to the f32 reference).
// x tiles are staged global->LDS with GLOBAL_LOAD_ASYNC_TO_LDS_B32 (ASYNCcnt)
// when the builtin is available, bypassing VGPRs entirely.

typedef float v2f __attribute__((ext_vector_type(2)));
typedef float v8f __attribute__((ext_vector_type(8)));

#define SEQ   512
#define INSZ  40
#define HID   64
#define HPAD  68   // h row stride in floats (pad to dodge bank conflicts)

#if defined(__has_builtin)
#  if __has_builtin(__builtin_amdgcn_tanhf)
#    define FAST_TANH(x) __builtin_amdgcn_tanhf(x)
#  else
#    define FAST_TANH(x) tanhf(x)
#  endif
#  if __has_builtin(__builtin_amdgcn_global_load_async_to_lds_b32)
#    define HAVE_ASYNC_LDS 1
#  endif
#else
#  define FAST_TANH(x) tanhf(x)
#endif

#ifdef HAVE_ASYNC_LDS
static __device__ __forceinline__ void async_copy_b32(const float* g, float* l)
{
    // Builtin signature (per hipcc diagnostic): generic int* params,
    // (global src, LDS dst, imm offset, imm cpol).
    __builtin_amdgcn_global_load_async_to_lds_b32((int*)g, (int*)l, 0, 0);
}
static __device__ __forceinline__ void async_wait0()
{
#if __has_builtin(__builtin_amdgcn_s_wait_asynccnt)
    __builtin_amdgcn_s_wait_asynccnt(0);
#else
    asm volatile("s_wait_asynccnt 0x0" ::: "memory");
#endif
}
#endif

__global__ __launch_bounds__(128)
void rnn_fused_kernel(const float* __restrict__ x,
                      const float* __restrict__ W_ih,
                      const float* __restrict__ W_hh,
                      const float* __restrict__ b_ih,
                      const float* __restrict__ b_hh,
                      const float* __restrict__ W_fc,
                      const float* __restrict__ b_fc,
                      float* __restrict__ out)
{
    __shared__ float hbuf[2][16 * HPAD];   // double-buffered hidden state tile
    __shared__ float xbuf[2][16 * INSZ];   // double-buffered x(t) tile

    const int tid  = threadIdx.x;
    const int wave = tid >> 5;          // 0..3 : 16-column tile of hidden dim
    const int lane = tid & 31;
    const int m    = lane & 15;         // row within tile (A-frag) / col N (B,C)
    const int kh   = lane >> 4;         // K-half selector in 16x16x4 layouts
    const int col0 = wave * 16;
    const int n    = col0 + m;          // this lane's output column (0..63)
    const int b0   = blockIdx.x * 16;   // batch rows of this block

    // ---- one-time register fragments ------------------------------------
    // B-frag of W_ih^T (K=40): B[k][n] = W_ih[n][k], lane holds k = 4kk+2kh+{0,1}
    v2f bih[10];
#pragma unroll
    for (int kk = 0; kk < 10; ++kk)
        bih[kk] = *(const v2f*)(W_ih + (size_t)n * INSZ + 4 * kk + 2 * kh);

    // B-frag of W_hh^T (K=64): B[k][n] = W_hh[n][k]
    v2f bhh[16];
#pragma unroll
    for (int kk = 0; kk < 16; ++kk)
        bhh[kk] = *(const v2f*)(W_hh + (size_t)n * HID + 4 * kk + 2 * kh);

    const float bias = b_ih[n] + b_hh[n];   // folded into C init
    const float bfc  = b_fc[0];

    // FC head: thread (om, os) reduces h[om][8*os .. 8*os+7]
    const int os = tid & 7;
    const int om = tid >> 3;
    float wfc[8];
#pragma unroll
    for (int j = 0; j < 8; ++j) wfc[j] = W_fc[os * 8 + j];

    // ---- staging slots: element i = tid + 128*j of the 16x40 tile --------
    // (row, col) per slot is constant over time; only the base moves by INSZ.
    int sofs[5];
    const float* gcur[5];
#pragma unroll
    for (int j = 0; j < 5; ++j) {
        const int i = tid + 128 * j;
        const int r = i / INSZ, cc = i % INSZ;
        sofs[j] = i;
        gcur[j] = x + (size_t)(b0 + r) * SEQ * INSZ + cc;   // t = 0 address
    }

    // ---- init: h(-1) = 0, stage x(0) ------------------------------------
    for (int i = tid; i < 16 * HPAD; i += 128) hbuf[0][i] = 0.0f;
#ifdef HAVE_ASYNC_LDS
#pragma unroll
    for (int j = 0; j < 5; ++j) async_copy_b32(gcur[j], &xbuf[0][sofs[j]]);
    async_wait0();
#else
#pragma unroll
    for (int j = 0; j < 5; ++j) xbuf[0][sofs[j]] = *gcur[j];
#endif
#pragma unroll
    for (int j = 0; j < 5; ++j) gcur[j] += INSZ;            // -> t = 1
    __syncthreads();

    // ---- recurrence ------------------------------------------------------
    for (int t = 0; t < SEQ; ++t) {
        const int p = t & 1;

        // stage x(t+1) into the other buffer (readers use xbuf[p])
        if (t + 1 < SEQ) {
#ifdef HAVE_ASYNC_LDS
#pragma unroll
            for (int j = 0; j < 5; ++j)
                async_copy_b32(gcur[j], &xbuf[p ^ 1][sofs[j]]);
#else
#pragma unroll
            for (int j = 0; j < 5; ++j)
                xbuf[p ^ 1][sofs[j]] = *gcur[j];
#endif
#pragma unroll
            for (int j = 0; j < 5; ++j) gcur[j] += INSZ;
        }
        // prefetch x(t+8) rows into cache (global_prefetch_b8)
        if (t + 8 < SEQ)
            __builtin_prefetch(&x[((size_t)(b0 + om) * SEQ + (t + 8)) * INSZ + os * 5], 0, 0);

        // FC head for h(t-1), read from the completed LDS buffer
        if (t > 0) {
            const float* hr = &hbuf[p][om * HPAD + os * 8];
            float s = 0.0f;
#pragma unroll
            for (int j = 0; j < 8; ++j) s += hr[j] * wfc[j];
            s += __shfl_xor(s, 1, 32);
            s += __shfl_xor(s, 2, 32);
            s += __shfl_xor(s, 4, 32);
            if (os == 0) out[(size_t)(b0 + om) * SEQ + (t - 1)] = s + bfc;
        }

        // C := bias (b_ih + b_hh broadcast along rows)
        v8f c;
#pragma unroll
        for (int r = 0; r < 8; ++r) c[r] = bias;

        // C += x(t)[16x40] @ W_ih^T[40x16]   (10 x K=4 WMMA)
#pragma unroll
        for (int kk = 0; kk < 10; ++kk) {
            const v2f a = *(const v2f*)(&xbuf[p][m * INSZ + 4 * kk + 2 * kh]);
            c = __builtin_amdgcn_wmma_f32_16x16x4_f32(
                    false, a, false, bih[kk], (short)0, c, false, false);
        }
        // C += h(t-1)[16x64] @ W_hh^T[64x16]  (16 x K=4 WMMA)
#pragma unroll
        for (int kk = 0; kk < 16; ++kk) {
            const v2f a = *(const v2f*)(&hbuf[p][m * HPAD + 4 * kk + 2 * kh]);
            c = __builtin_amdgcn_wmma_f32_16x16x4_f32(
                    false, a, false, bhh[kk], (short)0, c, false, false);
        }

        // h(t) = tanh(C); scatter C-layout tile to LDS (lane holds col n,
        // VGPR r holds row r + 8*kh)
#pragma unroll
        for (int r = 0; r < 8; ++r) {
            const float hv = FAST_TANH(c[r]);
            hbuf[p ^ 1][(r + 8 * kh) * HPAD + n] = hv;
        }
#ifdef HAVE_ASYNC_LDS
        async_wait0();      // x(t+1) tile resident before the barrier
#endif
        __syncthreads();
    }

    // FC head for the final step t = SEQ-1 (lives in hbuf[SEQ & 1])
    {
        const float* hr = &hbuf[SEQ & 1][om * HPAD + os * 8];
        float s = 0.0f;
#pragma unroll
        for (int j = 0; j < 8; ++j) s += hr[j] * wfc[j];
        s += __shfl_xor(s, 1, 32);
        s += __shfl_xor(s, 2, 32);
        s += __shfl_xor(s, 4, 32);
        if (os == 0) out[(size_t)(b0 + om) * SEQ + (SEQ - 1)] = s + bfc;
    }
}

extern "C" void kernel_launch(void* const* d_in, const int* in_sizes, int n_in,
                              void* d_out, int out_size, void* d_ws, size_t ws_size,
                              hipStream_t stream)
{
    (void)in_sizes; (void)n_in; (void)out_size; (void)d_ws; (void)ws_size;
    const float* x    = (const float*)d_in[0];
    const float* W_ih = (const float*)d_in[1];
    const float* W_hh = (const float*)d_in[2];
    const float* b_ih = (const float*)d_in[3];
    const float* b_hh = (const float*)d_in[4];
    const float* W_fc = (const float*)d_in[5];
    const float* b_fc = (const float*)d_in[6];
    float* out = (float*)d_out;

    dim3 grid(2048 / 16);   // 128 blocks, one 16-row batch tile each
    dim3 block(128);        // 4 wave32 per block (one per 16-col hidden tile)
    rnn_fused_kernel<<<grid, block, 0, stream>>>(x, W_ih, W_hh, b_ih, b_hh,
                                                 W_fc, b_fc, out);
}